// SA_Block_02_43576738185509
// MI455X (gfx1250) — compile-verified
//
#include <hip/hip_runtime.h>

typedef float v2f __attribute__((ext_vector_type(2)));
typedef float v8f __attribute__((ext_vector_type(8)));

#define BATCH   256
#define CH      64
#define NPTS    256
#define VD      64
#define NSTROKE 8

#define XSTR 260   // padded stride for 256-wide (N-major) rows
#define QSTR 68    // padded stride for 64-wide rows
#define NTHREADS 128
#define NWAVES 4

// ---- LDS layout (floats) ----
#define OFF_Q    0
#define OFF_KT   (OFF_Q  + NPTS*QSTR)          // q   : 256x68
#define OFF_V    (OFF_KT + CH*XSTR)            // kT  : 64x260
#define OFF_SCR  (OFF_V  + NPTS*QSTR)          // v   : 256x68
#define OFF_ZB   (OFF_SCR + CH*XSTR)           // scr : x (64x260) in P1 / 4 waves x (16x260) in P2
#define OFF_ZT   (OFF_ZB + NWAVES*NSTROKE*16)  // per-wave per-stroke row sums
#define OFF_RM   (OFF_ZT + NWAVES*16)          // per-wave row totals
#define OFF_BV   (OFF_RM + NWAVES*16)          // per-wave row maxima
#define OFF_INT  (OFF_BV + VD)                 // bias copy
#define SMEM_FLOATS (OFF_INT)
#define SMEM_BYTES  (SMEM_FLOATS*4 + (NPTS + 2*NSTROKE)*4)

static __device__ __forceinline__ v8f wmma_f32_16x16x4(v2f a, v2f b, v8f c) {
  // D = A(16x4,f32) * B(4x16,f32) + C(16x16,f32)   — exact fp32 matrix op
  return __builtin_amdgcn_wmma_f32_16x16x4_f32(false, a, false, b, (short)0, c,
                                               false, false);
}
static __device__ __forceinline__ v8f vzero8() {
  v8f z = {0.f, 0.f, 0.f, 0.f, 0.f, 0.f, 0.f, 0.f};
  return z;
}

__global__ void __launch_bounds__(NTHREADS)
sa_block_kernel(const float* __restrict__ x, const int* __restrict__ sidx,
                const float* __restrict__ Wq, const float* __restrict__ Wk,
                const float* __restrict__ Wv, const float* __restrict__ bvp,
                float* __restrict__ out) {
  extern __shared__ float smem[];
  float* sq   = smem + OFF_Q;    // q[n][j]    (n stride QSTR)
  float* skT  = smem + OFF_KT;   // kT[j][n]   (j stride XSTR)
  float* sv   = smem + OFF_V;    // v[n][j]    (n stride QSTR)
  float* scr  = smem + OFF_SCR;  // phase1: x[c][n]; phase2: per-wave e-tiles
  float* zbA  = smem + OFF_ZB;
  float* ztA  = smem + OFF_ZT;
  float* rmA  = smem + OFF_RM;
  float* bvs  = smem + OFF_BV;
  int*   ssid = (int*)(smem + OFF_INT);
  int*   sseg = ssid + NPTS;     // [0..7] = start, [8..15] = end

  const int tid  = threadIdx.x;
  const int lane = tid & 31;
  const int wv   = tid >> 5;
  const int l16  = lane & 15;
  const int hi   = lane >> 4;  // half of wave
  const int hsel = hi * 2;     // K offset for A/B operand halves
  const int h8   = hi * 8;     // M offset for C/D halves
  const int b    = blockIdx.x;

  // ---------- Phase 0: stage x, stroke ids, segment bounds, bias ----------
  for (int i = tid; i < CH * NPTS; i += NTHREADS) {
    int c = i >> 8, n = i & (NPTS - 1);
    scr[c * XSTR + n] = x[(size_t)b * CH * NPTS + i];
  }
  if (tid < NSTROKE) { sseg[tid] = NPTS; sseg[NSTROKE + tid] = 0; }
  if (tid < VD) bvs[tid] = bvp[tid];
  __syncthreads();
  for (int n = tid; n < NPTS; n += NTHREADS) {
    int s = sidx[b * NPTS + n];
    ssid[n] = s;
    atomicMin(&sseg[s], n);
    atomicMax(&sseg[NSTROKE + s], n + 1);
  }
  __syncthreads();

  // ---------- Phase 1: q = xT Wq^T, kT, v = xT Wv^T + bv  (WMMA fp32) ----------
  // 192 output tiles (3 mats * 16 n-tiles * 4 j-tiles) round-robined over waves.
  // Two independent accumulation chains (even/odd K chunks) for WMMA ILP:
  // only one wave runs per SIMD, so in-wave ILP is the sole latency cover.
  for (int t = wv; t < 192; t += NWAVES) {
    int which = t >> 6;
    int rem   = t & 63;
    int n0    = (rem >> 2) * 16;
    int j0    = (rem & 3) * 16;
    const float* W = (which == 0) ? Wq : ((which == 1) ? Wk : Wv);
    v8f d0 = vzero8(), d1 = vzero8();
#pragma unroll
    for (int kc = 0; kc < 16; kc += 2) {
      int c0 = kc * 4;
      v2f a0, b0, a1, b1;
      a0.x = scr[(c0 + hsel) * XSTR + n0 + l16];       // A[M=n][K=c]
      a0.y = scr[(c0 + 1 + hsel) * XSTR + n0 + l16];
      b0.x = W[(j0 + l16) * CH + c0 + hsel];           // B[K=c][N=j] = W[j][c]
      b0.y = W[(j0 + l16) * CH + c0 + 1 + hsel];
      a1.x = scr[(c0 + 4 + hsel) * XSTR + n0 + l16];
      a1.y = scr[(c0 + 5 + hsel) * XSTR + n0 + l16];
      b1.x = W[(j0 + l16) * CH + c0 + 4 + hsel];
      b1.y = W[(j0 + l16) * CH + c0 + 5 + hsel];
      d0 = wmma_f32_16x16x4(a0, b0, d0);
      d1 = wmma_f32_16x16x4(a1, b1, d1);
    }
    v8f d = d0 + d1;
    if (which == 0) {
#pragma unroll
      for (int vv = 0; vv < 8; ++vv) sq[(n0 + vv + h8) * QSTR + j0 + l16] = d[vv];
    } else if (which == 1) {
#pragma unroll
      for (int vv = 0; vv < 8; ++vv) skT[(j0 + l16) * XSTR + n0 + vv + h8] = d[vv];
    } else {
      float bj = bvs[j0 + l16];
#pragma unroll
      for (int vv = 0; vv < 8; ++vv) sv[(n0 + vv + h8) * QSTR + j0 + l16] = d[vv] + bj;
    }
  }
  __syncthreads();

  // ---------- Phase 2: per 16-query tile: scores -> exp -> 2-GEMM attention ----------
  float* eb  = scr + wv * (16 * XSTR);
  float* zb  = zbA + wv * (NSTROKE * 16);
  float* zt  = ztA + wv * 16;
  float* rmw = rmA + wv * 16;
  const float c1 = __expf(1e-5f);  // out-of-stroke mask factor (mask*10)

  for (int qt = wv; qt < 16; qt += NWAVES) {
    int n0 = qt * 16;

    // --- scores tile (16 x 256), scaled by 10/sqrt(QK) = 1.25 ---
    float rm[8];
#pragma unroll
    for (int vv = 0; vv < 8; ++vv) rm[vv] = -3.0e38f;
    for (int kt = 0; kt < 16; ++kt) {
      v8f d0 = vzero8(), d1 = vzero8();
#pragma unroll
      for (int kc = 0; kc < 16; kc += 2) {
        int c0 = kc * 4;
        v2f a0, b0, a1, b1;
        a0.x = sq[(n0 + l16) * QSTR + c0 + hsel];
        a0.y = sq[(n0 + l16) * QSTR + c0 + 1 + hsel];
        b0.x = skT[(c0 + hsel) * XSTR + kt * 16 + l16];
        b0.y = skT[(c0 + 1 + hsel) * XSTR + kt * 16 + l16];
        a1.x = sq[(n0 + l16) * QSTR + c0 + 4 + hsel];
        a1.y = sq[(n0 + l16) * QSTR + c0 + 5 + hsel];
        b1.x = skT[(c0 + 4 + hsel) * XSTR + kt * 16 + l16];
        b1.y = skT[(c0 + 5 + hsel) * XSTR + kt * 16 + l16];
        d0 = wmma_f32_16x16x4(a0, b0, d0);
        d1 = wmma_f32_16x16x4(a1, b1, d1);
      }
      v8f d = d0 + d1;
#pragma unroll
      for (int vv = 0; vv < 8; ++vv) {
        float s10 = d[vv] * 1.25f;
        eb[(vv + h8) * XSTR + kt * 16 + l16] = s10;
        rm[vv] = fmaxf(rm[vv], s10);
      }
    }
    // row maxima: reduce 16 lanes within each half, publish per-row to LDS
#pragma unroll
    for (int vv = 0; vv < 8; ++vv) {
      float m = rm[vv];
#pragma unroll
      for (int off = 8; off > 0; off >>= 1) m = fmaxf(m, __shfl_xor(m, off, 16));
      if (l16 == 0) rmw[vv + h8] = m;
    }
    __syncthreads();

    // --- exp in place + total and per-stroke (contiguous segment) row sums ---
    if (lane < 16) {
      const float rmr = rmw[lane];
      float* erow = eb + lane * XSTR;
#pragma unroll
      for (int i = 0; i < NSTROKE; ++i) zb[i * 16 + lane] = 0.0f;
      float acc = 0.0f, tot = 0.0f;
      int cur = ssid[0];
      for (int k = 0; k < NPTS; ++k) {
        float e = __expf(erow[k] - rmr);
        erow[k] = e;
        int sk = ssid[k];
        if (sk != cur) { zb[cur * 16 + lane] = acc; acc = 0.0f; cur = sk; }
        acc += e;
        tot += e;
      }
      zb[cur * 16 + lane] = acc;
      zt[lane] = tot;
    }
    __syncthreads();

    // --- full GEMM: A = e @ v  (16 x 64), K = 256 keys; 4 independent chains ---
    v8f accA[4];
#pragma unroll
    for (int nt = 0; nt < 4; ++nt) accA[nt] = vzero8();
    for (int ks = 0; ks < 64; ++ks) {
      int k0 = ks * 4;
      v2f a;
      a.x = eb[l16 * XSTR + k0 + hsel];
      a.y = eb[l16 * XSTR + k0 + 1 + hsel];
#pragma unroll
      for (int nt = 0; nt < 4; ++nt) {
        v2f bb;
        bb.x = sv[(k0 + hsel) * QSTR + nt * 16 + l16];
        bb.y = sv[(k0 + 1 + hsel) * QSTR + nt * 16 + l16];
        accA[nt] = wmma_f32_16x16x4(a, bb, accA[nt]);
      }
    }

    // --- per-stroke: out += [c1*A + (1-c1)*P_i] / Z_i,  Z_i = c1*ztot + (1-c1)*z_i ---
    v8f outv[4];
#pragma unroll
    for (int nt = 0; nt < 4; ++nt) outv[nt] = vzero8();
    float sumInv[8];
#pragma unroll
    for (int vv = 0; vv < 8; ++vv) sumInv[vv] = 0.0f;

    for (int i = 0; i < NSTROKE; ++i) {
      float invZ[8];
#pragma unroll
      for (int vv = 0; vv < 8; ++vv) {
        float zi  = zb[i * 16 + vv + h8];
        float ztv = zt[vv + h8];
        invZ[vv] = 1.0f / (c1 * ztv + (1.0f - c1) * zi);
        sumInv[vv] += invZ[vv];
      }
      int st = sseg[i], en = sseg[NSTROKE + i];
      if (st < en) {  // wave-uniform branch; EXEC stays all-ones for WMMA
        v8f P[4];
#pragma unroll
        for (int nt = 0; nt < 4; ++nt) P[nt] = vzero8();
        for (int k0 = (st & ~3); k0 < en; k0 += 4) {
          int ka = k0 + hsel;
          v2f a;
          a.x = (ka     >= st && ka     < en) ? eb[l16 * XSTR + ka]     : 0.0f;
          a.y = (ka + 1 >= st && ka + 1 < en) ? eb[l16 * XSTR + ka + 1] : 0.0f;
#pragma unroll
          for (int nt = 0; nt < 4; ++nt) {
            v2f bb;
            bb.x = sv[ka * QSTR + nt * 16 + l16];   // ka <= 255: in-bounds
            bb.y = sv[(ka + 1) * QSTR + nt * 16 + l16];
            P[nt] = wmma_f32_16x16x4(a, bb, P[nt]);
          }
        }
#pragma unroll
        for (int nt = 0; nt < 4; ++nt)
#pragma unroll
          for (int vv = 0; vv < 8; ++vv)
            outv[nt][vv] += (1.0f - c1) * invZ[vv] * P[nt][vv];
      }
    }
#pragma unroll
    for (int nt = 0; nt < 4; ++nt)
#pragma unroll
      for (int vv = 0; vv < 8; ++vv)
        outv[nt][vv] += c1 * sumInv[vv] * accA[nt][vv];

    // --- store: out (B, V, N) ---
#pragma unroll
    for (int nt = 0; nt < 4; ++nt)
#pragma unroll
      for (int vv = 0; vv < 8; ++vv)
        out[(size_t)b * VD * NPTS + (nt * 16 + l16) * NPTS + n0 + vv + h8] =
            outv[nt][vv];
    __syncthreads();
  }
}

extern "C" void kernel_launch(void* const* d_in, const int* in_sizes, int n_in,
                              void* d_out, int out_size, void* d_ws, size_t ws_size,
                              hipStream_t stream) {
  (void)in_sizes; (void)n_in; (void)d_ws; (void)ws_size; (void)out_size;
  const float* x   = (const float*)d_in[0];
  const int*  sidx = (const int*)d_in[1];
  const float* Wq  = (const float*)d_in[2];
  const float* Wk  = (const float*)d_in[3];
  const float* Wv  = (const float*)d_in[4];
  const float* bv  = (const float*)d_in[5];
  float* out = (float*)d_out;

  hipFuncSetAttribute(reinterpret_cast<const void*>(sa_block_kernel),
                      hipFuncAttributeMaxDynamicSharedMemorySize, SMEM_BYTES);
  sa_block_kernel<<<BATCH, NTHREADS, SMEM_BYTES, stream>>>(x, sidx, Wq, Wk, Wv,
                                                           bv, out);
}